// AEGConv2d_46909632806927
// MI455X (gfx1250) — compile-verified
//
#include <hip/hip_runtime.h>

typedef __attribute__((ext_vector_type(2))) float v2f;
typedef __attribute__((ext_vector_type(8))) float v8f;

#define NN    8
#define CIN   32
#define HH    64
#define WW    64
#define COUT  64
#define KDIM  288   // 32 ic * 9 taps
#define KCH   72    // K-chunk: 8 ic * 9
#define NKCH  4     // 4 chunks of 72 = 288

// ---------------------------------------------------------------------------
// Kernel 1: precompute V[parity][oc][ic*9+m] = weight-only factor of each
// bilinear term of the unrolled AEG recurrence.  2*64*32 threads, 9 outs each.
// ---------------------------------------------------------------------------
__global__ void aeg_prep_v(const float* __restrict__ w, float* __restrict__ V) {
    int t = blockIdx.x * blockDim.x + threadIdx.x;      // 0 .. 4095
    if (t >= 2 * COUT * CIN) return;
    int c   = t >> 11;          // parity class
    int rem = t & 2047;
    int oc  = rem >> 5;
    int ic  = rem & 31;
    const float* y = w + (oc * CIN + ic) * 9;
    float y0=y[0], y1=y[1], y2=y[2], y3=y[3], y4=y[4],
          y5=y[5], y6=y[6], y7=y[7], y8=y[8];
    float o[9];
    if (c == 0) {   // (i+j) even: b_l = y for even l -> even-y suffix products
        float Q8 = y8, Q6 = y6*Q8, Q4 = y4*Q6, Q2 = y2*Q4, Q0 = y0*Q2;
        o[0]=Q0;    o[1]=y1*Q2; o[2]=Q2;    o[3]=y3*Q4; o[4]=Q4;
        o[5]=y5*Q6; o[6]=Q6;    o[7]=y7*Q8; o[8]=Q8;
    } else {        // (i+j) odd: b_l = y for odd l -> odd-y suffix products
        float S7 = y7, S5 = y5*S7, S3 = y3*S5, S1 = y1*S3;
        o[0]=y0*S1; o[1]=S1;    o[2]=y2*S3; o[3]=S3;    o[4]=y4*S5;
        o[5]=S5;    o[6]=y6*S7; o[7]=S7;    o[8]=y8;
    }
    float* dst = V + (c * COUT + oc) * KDIM + ic * 9;
#pragma unroll
    for (int m = 0; m < 9; ++m) dst[m] = o[m];
}

// ---------------------------------------------------------------------------
// Kernel 2: out[pos, oc] = U[pos, 0:288] * V[parity, 0:288, oc] via
// V_WMMA_F32_16X16X4_F32.  8 waves/block; wave = one 16-position M-tile
// (parity-uniform checkerboard), 4 oc N-tiles.  K in 4 chunks of 72.
// ---------------------------------------------------------------------------
__launch_bounds__(256)
__global__ void aeg_wmma(const float* __restrict__ x,
                         const float* __restrict__ V,
                         float* __restrict__ out) {
    __shared__ float sV[COUT * KCH];        // 64*72  floats = 18 KB (per chunk)
    __shared__ float sU[8 * 16 * KCH];      // 8w*16p*72 floats = 36 KB

    const int tid  = threadIdx.x;
    const int wave = tid >> 5;
    const int lane = tid & 31;
    const int lo   = lane & 15;
    const int hi   = lane >> 4;

    const int tile0  = blockIdx.x * 8;          // 2048 M-tiles total
    const int c      = (tile0 >> 10) & 1;       // tiles [0,1024) even, [1024,2048) odd
    const int mytile = tile0 + wave;
    const int e_base = (mytile & 1023) * 16;    // position index within class

    v8f acc0 = {}, acc1 = {}, acc2 = {}, acc3 = {};

    for (int kc = 0; kc < NKCH; ++kc) {
        // ---- stage V chunk: sV[oc*72 + kk], linear cooperative load --------
        const int vbase = c * COUT * KDIM + kc * KCH;
#pragma unroll
        for (int s = 0; s < 18; ++s) {
            int idx = tid + 256 * s;            // 0 .. 4607  (= oc*72 + kk)
            int oc  = idx / KCH;
            int kk  = idx - oc * KCH;
            sV[idx] = V[vbase + oc * KDIM + kk];
        }
        // ---- compute U chunk: 4 (pos, ic) pairs per lane -------------------
#pragma unroll
        for (int p = 0; p < 4; ++p) {
            int idx = lane + 32 * p;            // 0 .. 127
            int pos = idx >> 3;
            int icg = idx & 7;
            int ic  = kc * 8 + icg;
            int e   = e_base + pos;
            int n   = e >> 11;
            int r2  = e & 2047;
            int i   = r2 >> 5;
            int jh  = r2 & 31;
            int j   = 2 * jh + ((i + c) & 1);
            const float* xb = x + (n * CIN + ic) * (HH * WW);
            float xv[9];
#pragma unroll
            for (int dh = 0; dh < 3; ++dh)
#pragma unroll
                for (int dw = 0; dw < 3; ++dw) {
                    int ii = i + dh - 1, jj = j + dw - 1;
                    bool ok = ((unsigned)ii < (unsigned)HH) &&
                              ((unsigned)jj < (unsigned)WW);
                    xv[dh * 3 + dw] = ok ? xb[ii * WW + jj] : 0.0f;
                }
            float u[9];
            if (c == 0) {   // even: odd-x suffix products
                float P7 = xv[7], P5 = xv[5]*P7, P3 = xv[3]*P5, P1 = xv[1]*P3;
                u[0]=xv[0]*P1; u[1]=P1;       u[2]=xv[2]*P3; u[3]=P3;
                u[4]=xv[4]*P5; u[5]=P5;       u[6]=xv[6]*P7; u[7]=P7; u[8]=xv[8];
            } else {        // odd: even-x suffix products
                float R8 = xv[8], R6 = xv[6]*R8, R4 = xv[4]*R6,
                      R2 = xv[2]*R4, R0 = xv[0]*R2;
                u[0]=R0;       u[1]=xv[1]*R2; u[2]=R2;       u[3]=xv[3]*R4;
                u[4]=R4;       u[5]=xv[5]*R6; u[6]=R6;       u[7]=xv[7]*R8; u[8]=R8;
            }
            float* du = &sU[(wave * 16 + pos) * KCH + icg * 9];
#pragma unroll
            for (int m = 0; m < 9; ++m) du[m] = u[m];
        }
        __syncthreads();

        // ---- WMMA sweep over this K chunk ---------------------------------
        const float* uB = &sU[wave * 16 * KCH];
        for (int k0 = 0; k0 < KCH; k0 += 4) {
            // A frag: lane(lo,hi) holds A[M=lo][k0+2hi .. +1]  (8B aligned)
            v2f a  = *(const v2f*)&uB[lo * KCH + k0 + 2 * hi];
            // B frags: lane(lo,hi) holds B[k0+2hi .. +1][N=octile*16+lo]
            v2f b0 = *(const v2f*)&sV[(lo)      * KCH + k0 + 2 * hi];
            v2f b1 = *(const v2f*)&sV[(16 + lo) * KCH + k0 + 2 * hi];
            v2f b2 = *(const v2f*)&sV[(32 + lo) * KCH + k0 + 2 * hi];
            v2f b3 = *(const v2f*)&sV[(48 + lo) * KCH + k0 + 2 * hi];
            acc0 = __builtin_amdgcn_wmma_f32_16x16x4_f32(false, a, false, b0,
                       (short)0, acc0, false, false);
            acc1 = __builtin_amdgcn_wmma_f32_16x16x4_f32(false, a, false, b1,
                       (short)0, acc1, false, false);
            acc2 = __builtin_amdgcn_wmma_f32_16x16x4_f32(false, a, false, b2,
                       (short)0, acc2, false, false);
            acc3 = __builtin_amdgcn_wmma_f32_16x16x4_f32(false, a, false, b3,
                       (short)0, acc3, false, false);
        }
        __syncthreads();
    }

    // ---- store: D layout VGPR r, half hi -> row M = r + 8*hi, col N = lo ---
#pragma unroll
    for (int t = 0; t < 4; ++t) {
        v8f A = (t == 0) ? acc0 : (t == 1) ? acc1 : (t == 2) ? acc2 : acc3;
        int oc = t * 16 + lo;
#pragma unroll
        for (int r = 0; r < 8; ++r) {
            int M  = r + 8 * hi;
            int e  = e_base + M;
            int n  = e >> 11;
            int r2 = e & 2047;
            int i  = r2 >> 5;
            int jh = r2 & 31;
            int j  = 2 * jh + ((i + c) & 1);
            out[((n * COUT + oc) * HH + i) * WW + j] = A[r];
        }
    }
}

extern "C" void kernel_launch(void* const* d_in, const int* in_sizes, int n_in,
                              void* d_out, int out_size, void* d_ws, size_t ws_size,
                              hipStream_t stream) {
    const float* x = (const float*)d_in[0];      // (8,32,64,64) f32
    const float* w = (const float*)d_in[1];      // (64,32,3,3)  f32
    float* out = (float*)d_out;                  // (8,64,64,64) f32
    float* V   = (float*)d_ws;                   // 2*64*288 f32 = 144 KB

    aeg_prep_v<<<16, 256, 0, stream>>>(w, V);
    aeg_wmma<<<256, 256, 0, stream>>>(x, V, out);
}